// PointNetShapeServoKp_32014686224759
// MI455X (gfx1250) — compile-verified
//
#include <hip/hip_runtime.h>
#include <math.h>

typedef __attribute__((ext_vector_type(2))) float v2f;
typedef __attribute__((ext_vector_type(8))) float v8f;

#define BB   32      // 16 batches of xyz + 16 of xyz_goal (shared weights)
#define NPTS 2048
#define NP1  64      // npoint for SA1
#define NS1  8       // nsample for SA1

struct WnP {  // weightnet params (3 -> 8 -> 8 -> 16)
  const float *w0,*b0,*g0,*be0;
  const float *w1,*b1,*g1,*be1;
  const float *w2,*b2,*g2,*be2;
};

__device__ __forceinline__ const float* pc_base(const float* xyz, const float* goal,
                                                int b, int N) {
  return ((b < 16) ? xyz : goal) + (size_t)(b & 15) * 6 * N;
}

// ---------------------------------------------------------------- sqnorm
__global__ void sqnorm_kernel(const float* __restrict__ xyz, const float* __restrict__ goal,
                              float* __restrict__ sqn, int N) {
  int i = blockIdx.x * blockDim.x + threadIdx.x;
  if (i >= BB * N) return;
  int b = i / N, p = i % N;
  const float* base = pc_base(xyz, goal, b, N);
  float x = base[p], y = base[N + p], z = base[2 * N + p];
  sqn[i] = x * x + y * y + z * z;
}

// ------------------------------------------------- WMMA pairwise density
// One wave per 16-row strip; loops over 2048/16 column tiles.
// d2 = |xi|^2 + |xj|^2 - 2*dot(xi,xj); dot via V_WMMA_F32_16X16X4_F32 (K=3 pad 4).
__global__ void density_kernel(const float* __restrict__ xyz, const float* __restrict__ goal,
                               const float* __restrict__ sqn, float* __restrict__ invd,
                               int N, float bw) {
  int nstrip = N >> 4;
  int strip = blockIdx.x;
  int b = strip / nstrip;
  int i0 = (strip % nstrip) << 4;
  const float* base = pc_base(xyz, goal, b, N);
  int lane = threadIdx.x & 31;
  int half = lane >> 4;          // 0: K=0,1 ; 1: K=2,3
  int l15  = lane & 15;
  int kh   = half * 2;

  // A fragment: rows i0..i0+15, coords K (zero-pad K=3)
  v2f a;
  a.x = (kh + 0 < 3) ? base[(kh + 0) * N + i0 + l15] : 0.f;
  a.y = (kh + 1 < 3) ? base[(kh + 1) * N + i0 + l15] : 0.f;

  // row sqnorms for this lane's result rows (VGPR r -> row r + half*8)
  int rbase = half * 8;
  float ra[8];
#pragma unroll
  for (int r = 0; r < 8; ++r) ra[r] = sqn[b * N + i0 + rbase + r];

  float accs[8];
#pragma unroll
  for (int r = 0; r < 8; ++r) accs[r] = 0.f;

  const float inv2b2 = 1.f / (2.f * bw * bw);
  for (int j0 = 0; j0 < N; j0 += 16) {
    v2f bf;
    bf.x = (kh + 0 < 3) ? base[(kh + 0) * N + j0 + l15] : 0.f;
    bf.y = (kh + 1 < 3) ? base[(kh + 1) * N + j0 + l15] : 0.f;
    v8f c = {};
    c = __builtin_amdgcn_wmma_f32_16x16x4_f32(false, a, false, bf,
                                              (short)0, c, false, false);
    float cb = sqn[b * N + j0 + l15];   // column sqnorm for this lane's column
#pragma unroll
    for (int r = 0; r < 8; ++r) {
      float d2 = ra[r] + cb - 2.f * c[r];
      accs[r] += __expf(-d2 * inv2b2);
    }
  }
  // reduce across the 16 lanes of each half (offsets <16 stay within group)
#pragma unroll
  for (int r = 0; r < 8; ++r) {
    float v = accs[r];
    v += __shfl_xor(v, 1, 32);
    v += __shfl_xor(v, 2, 32);
    v += __shfl_xor(v, 4, 32);
    v += __shfl_xor(v, 8, 32);
    accs[r] = v;
  }
  if (l15 == 0) {
    float norm = 2.5f * bw * (float)N;   // inv_density = 1/mean(g)
#pragma unroll
    for (int r = 0; r < 8; ++r)
      invd[b * N + i0 + rbase + r] = norm / accs[r];
  }
}

// ---------------------------------------------------------------- FPS
__global__ void fps_kernel(const float* __restrict__ xyz, const float* __restrict__ goal,
                           int* __restrict__ fpsi, int N, int npoint) {
  __shared__ float dist[NPTS];
  __shared__ float redv[256];
  __shared__ int   redi[256];
  __shared__ int   curf;
  int b = blockIdx.x;
  const float* base = pc_base(xyz, goal, b, N);
  int t = threadIdx.x;
  for (int i = t; i < N; i += 256) dist[i] = 1e10f;
  if (t == 0) curf = 0;
  __syncthreads();
  for (int it = 0; it < npoint; ++it) {
    int far = curf;
    if (t == 0) fpsi[b * npoint + it] = far;    // emit carry (pre-update), matches scan
    float cx = base[far], cy = base[N + far], cz = base[2 * N + far];
    float best = -1.f; int bi = N;
    for (int i = t; i < N; i += 256) {
      float dx = base[i] - cx, dy = base[N + i] - cy, dz = base[2 * N + i] - cz;
      float d = dx * dx + dy * dy + dz * dz;
      float nd = fminf(dist[i], d);
      dist[i] = nd;
      if (nd > best) { best = nd; bi = i; }     // first-max within ascending stride
    }
    redv[t] = best; redi[t] = bi;
    __syncthreads();
    for (int s = 128; s > 0; s >>= 1) {
      if (t < s) {
        if (redv[t + s] > redv[t] ||
            (redv[t + s] == redv[t] && redi[t + s] < redi[t])) {
          redv[t] = redv[t + s]; redi[t] = redi[t + s];
        }
      }
      __syncthreads();
    }
    if (t == 0) curf = redi[0];
    __syncthreads();
  }
}

// ---------------------------------------------------------------- kNN (k=8)
__global__ void knn_kernel(const float* __restrict__ xyz, const float* __restrict__ goal,
                           const int* __restrict__ fpsi, int* __restrict__ knni,
                           float* __restrict__ l1x, int N) {
  __shared__ float sd[64 * 8];
  __shared__ int   si[64 * 8];
  int bs = blockIdx.x;
  int b = bs >> 6;
  const float* base = pc_base(xyz, goal, b, N);
  int t = threadIdx.x;               // 64 threads
  int fi = fpsi[bs];
  float qx = base[fi], qy = base[N + fi], qz = base[2 * N + fi];
  if (t == 0) { l1x[bs * 3 + 0] = qx; l1x[bs * 3 + 1] = qy; l1x[bs * 3 + 2] = qz; }

  float bd[8]; int bi[8];
#pragma unroll
  for (int k = 0; k < 8; ++k) { bd[k] = 3.0e38f; bi[k] = 0x7fffffff; }
  for (int i = t; i < N; i += 64) {
    float dx = base[i] - qx, dy = base[N + i] - qy, dz = base[2 * N + i] - qz;
    float d = dx * dx + dy * dy + dz * dz;
    if (d < bd[7] || (d == bd[7] && i < bi[7])) {
      int pos = 7;
      while (pos > 0 && (d < bd[pos - 1] || (d == bd[pos - 1] && i < bi[pos - 1]))) --pos;
      for (int q = 7; q > pos; --q) { bd[q] = bd[q - 1]; bi[q] = bi[q - 1]; }
      bd[pos] = d; bi[pos] = i;
    }
  }
#pragma unroll
  for (int k = 0; k < 8; ++k) { sd[t * 8 + k] = bd[k]; si[t * 8 + k] = bi[k]; }
  __syncthreads();
  if (t == 0) {
    for (int k = 0; k < 8; ++k) {
      float md = 3.0e38f; int mi = 0x7fffffff, mp = 0;
      for (int q = 0; q < 64 * 8; ++q) {
        if (sd[q] < md || (sd[q] == md && si[q] < mi)) { md = sd[q]; mi = si[q]; mp = q; }
      }
      sd[mp] = 3.3e38f;
      knni[bs * 8 + k] = mi;
    }
  }
}

// --------------------------- SA1: gather + MLP + density scale + weightnet + einsum
__global__ void sa1_group_kernel(const float* __restrict__ xyz, const float* __restrict__ goal,
                                 const int* __restrict__ knni, const float* __restrict__ invd,
                                 const float* __restrict__ l1x,
                                 const float* __restrict__ mlp_w, const float* __restrict__ mlp_b,
                                 const float* __restrict__ mlp_g, const float* __restrict__ mlp_be,
                                 WnP wn, float* __restrict__ A1, int N) {
  __shared__ float feat[NS1][9];
  __shared__ float xn[NS1][3];
  __shared__ float wf[16][NS1];
  __shared__ float gd[NS1];
  __shared__ float scale[NS1];
  int bs = blockIdx.x;
  int b = bs >> 6;
  const float* base = pc_base(xyz, goal, b, N);
  int t = threadIdx.x;               // 64 threads = out channels
  if (t < NS1) {
    int p = knni[bs * 8 + t];
    float nx = base[p]         - l1x[bs * 3 + 0];
    float ny = base[N + p]     - l1x[bs * 3 + 1];
    float nz = base[2 * N + p] - l1x[bs * 3 + 2];
    xn[t][0] = nx; xn[t][1] = ny; xn[t][2] = nz;
    feat[t][0] = nx; feat[t][1] = ny; feat[t][2] = nz;
    for (int c = 0; c < 6; ++c) feat[t][3 + c] = base[c * N + p];
    gd[t] = invd[b * N + p];
  }
  __syncthreads();
  if (t == 0) {
    float mx = gd[0];
    for (int k = 1; k < NS1; ++k) mx = fmaxf(mx, gd[k]);
    for (int k = 0; k < NS1; ++k) scale[k] = gd[k] / mx;
  }
  __syncthreads();
  // MLP 9 -> 64, bn+relu, density scale
  float x[NS1];
#pragma unroll
  for (int k = 0; k < NS1; ++k) {
    float acc = mlp_b[t];
    for (int c = 0; c < 9; ++c) acc += mlp_w[t * 9 + c] * feat[k][c];
    acc = fmaxf(acc * mlp_g[t] + mlp_be[t], 0.f);
    x[k] = acc * scale[k];
  }
  // weightnet per sample (threads 0..7)
  if (t < NS1) {
    float v0[8], v1[8];
    for (int o = 0; o < 8; ++o) {
      float a = wn.b0[o];
      for (int c = 0; c < 3; ++c) a += wn.w0[o * 3 + c] * xn[t][c];
      v0[o] = fmaxf(a * wn.g0[o] + wn.be0[o], 0.f);
    }
    for (int o = 0; o < 8; ++o) {
      float a = wn.b1[o];
      for (int c = 0; c < 8; ++c) a += wn.w1[o * 8 + c] * v0[c];
      v1[o] = fmaxf(a * wn.g1[o] + wn.be1[o], 0.f);
    }
    for (int o = 0; o < 16; ++o) {
      float a = wn.b2[o];
      for (int c = 0; c < 8; ++c) a += wn.w2[o * 8 + c] * v1[c];
      wf[o][t] = fmaxf(a * wn.g2[o] + wn.be2[o], 0.f);
    }
  }
  __syncthreads();
  // einsum bcks,bwks->bscw ; flatten (c,w) -> c*16+w
  float* outp = A1 + (long long)bs * 1024 + t * 16;
#pragma unroll
  for (int w = 0; w < 16; ++w) {
    float acc = 0.f;
#pragma unroll
    for (int k = 0; k < NS1; ++k) acc += x[k] * wf[w][k];
    outp[w] = acc;
  }
}

// ----------------------------------- generic WMMA GEMM: C = A*B^T (+bias)(*g+be)(relu)
// A: M x K row-major (per-batch stride strideA), B: N x K row-major (shared weights).
__global__ void gemm_wmma_kernel(const float* __restrict__ A, const float* __restrict__ Bw,
                                 const float* __restrict__ bias, const float* __restrict__ gam,
                                 const float* __restrict__ bet, float* __restrict__ C,
                                 int M, int N, int K,
                                 long long strideA, long long strideC, int relu) {
  int n0 = blockIdx.x << 4;
  int m0 = blockIdx.y << 4;
  const float* Ab = A + (long long)blockIdx.z * strideA;
  float* Cb = C + (long long)blockIdx.z * strideC;
  int lane = threadIdx.x & 31;
  int half = lane >> 4;
  int l15  = lane & 15;
  int m = m0 + l15;
  int n = n0 + l15;
  bool mok = (m < M);
  bool nok = (n < N);
  v8f acc = {};
  for (int k0 = 0; k0 < K; k0 += 4) {
    int ks = k0 + half * 2;
    v2f a, bf;
    a.x  = mok ? Ab[(long long)m * K + ks]     : 0.f;
    a.y  = mok ? Ab[(long long)m * K + ks + 1] : 0.f;
    bf.x = nok ? Bw[(long long)n * K + ks]     : 0.f;
    bf.y = nok ? Bw[(long long)n * K + ks + 1] : 0.f;
    acc = __builtin_amdgcn_wmma_f32_16x16x4_f32(false, a, false, bf,
                                                (short)0, acc, false, false);
  }
  int rowb = m0 + half * 8;
#pragma unroll
  for (int r = 0; r < 8; ++r) {
    int row = rowb + r, col = n0 + l15;
    if (row < M && col < N) {
      float v = acc[r];
      if (bias) v += bias[col];
      if (gam)  v = v * gam[col] + bet[col];
      if (relu) v = fmaxf(v, 0.f);
      Cb[(long long)row * N + col] = v;
    }
  }
}

// ------------------------------------------------ SA3 density (64 pts, bw=0.4)
__global__ void density3_kernel(const float* __restrict__ l1x, float* __restrict__ invd3) {
  __shared__ float px[NP1], py[NP1], pz[NP1];
  int b = blockIdx.x, t = threadIdx.x;   // 64 threads
  px[t] = l1x[(b * NP1 + t) * 3 + 0];
  py[t] = l1x[(b * NP1 + t) * 3 + 1];
  pz[t] = l1x[(b * NP1 + t) * 3 + 2];
  __syncthreads();
  const float inv2b2 = 1.f / (2.f * 0.4f * 0.4f);
  float acc = 0.f;
  for (int j = 0; j < NP1; ++j) {
    float dx = px[t] - px[j], dy = py[t] - py[j], dz = pz[t] - pz[j];
    acc += __expf(-(dx * dx + dy * dy + dz * dz) * inv2b2);
  }
  invd3[b * NP1 + t] = (2.5f * 0.4f * (float)NP1) / acc;   // = 64/acc
}

// ---------------------------------- SA3: concat + MLP(67->128) + scale + weightnet + einsum
__global__ void sa3_kernel(const float* __restrict__ l1x, const float* __restrict__ l1p,
                           const float* __restrict__ invd3,
                           const float* __restrict__ mlp_w, const float* __restrict__ mlp_b,
                           const float* __restrict__ mlp_g, const float* __restrict__ mlp_be,
                           WnP wn, float* __restrict__ A3) {
  __shared__ float feat[NP1][67];
  __shared__ float wf[16][NP1];
  __shared__ float scale[NP1];
  __shared__ float xsh[128][NP1];
  int b = blockIdx.x, t = threadIdx.x;   // 128 threads
  for (int i = t; i < NP1 * 67; i += 128) {
    int k = i / 67, c = i % 67;
    feat[k][c] = (c < 3) ? l1x[(b * NP1 + k) * 3 + c]
                         : l1p[(b * NP1 + k) * 64 + (c - 3)];
  }
  if (t < NP1) scale[t] = invd3[b * NP1 + t];
  __syncthreads();
  if (t == 0) {
    float mx = scale[0];
    for (int k = 1; k < NP1; ++k) mx = fmaxf(mx, scale[k]);
    for (int k = 0; k < NP1; ++k) scale[k] /= mx;
  }
  __syncthreads();
  for (int k = 0; k < NP1; ++k) {
    float acc = mlp_b[t];
    for (int c = 0; c < 67; ++c) acc += mlp_w[t * 67 + c] * feat[k][c];
    acc = fmaxf(acc * mlp_g[t] + mlp_be[t], 0.f);
    xsh[t][k] = acc * scale[k];
  }
  if (t < NP1) {
    float v0[8], v1[8];
    for (int o = 0; o < 8; ++o) {
      float a = wn.b0[o];
      for (int c = 0; c < 3; ++c) a += wn.w0[o * 3 + c] * feat[t][c];
      v0[o] = fmaxf(a * wn.g0[o] + wn.be0[o], 0.f);
    }
    for (int o = 0; o < 8; ++o) {
      float a = wn.b1[o];
      for (int c = 0; c < 8; ++c) a += wn.w1[o * 8 + c] * v0[c];
      v1[o] = fmaxf(a * wn.g1[o] + wn.be1[o], 0.f);
    }
    for (int o = 0; o < 16; ++o) {
      float a = wn.b2[o];
      for (int c = 0; c < 8; ++c) a += wn.w2[o * 8 + c] * v1[c];
      wf[o][t] = fmaxf(a * wn.g2[o] + wn.be2[o], 0.f);
    }
  }
  __syncthreads();
  float* outp = A3 + (long long)b * 2048 + t * 16;
#pragma unroll
  for (int w = 0; w < 16; ++w) {
    float acc = 0.f;
    for (int k = 0; k < NP1; ++k) acc += xsh[t][k] * wf[w][k];
    outp[w] = acc;
  }
}

// ---------------------------------------------------------------- head helpers
__global__ void diff_kernel(const float* __restrict__ l3, float* __restrict__ h) {
  int i = blockIdx.x * blockDim.x + threadIdx.x;   // 16*128
  if (i < 16 * 128) {
    int r = i >> 7, c = i & 127;
    h[i] = l3[(16 + r) * 128 + c] - l3[r * 128 + c];   // g - x
  }
}

__global__ void groupnorm_relu_kernel(float* __restrict__ x, const float* __restrict__ g,
                                      const float* __restrict__ bta, int C) {
  __shared__ float buf[64];
  __shared__ float mv[2];
  int row = blockIdx.x, t = threadIdx.x;   // 64 threads
  float v = (t < C) ? x[row * C + t] : 0.f;
  if (t < 64) buf[t] = v;
  __syncthreads();
  if (t == 0) {
    float m = 0.f;
    for (int i = 0; i < C; ++i) m += buf[i];
    m /= (float)C;
    float var = 0.f;
    for (int i = 0; i < C; ++i) { float d = buf[i] - m; var += d * d; }
    var /= (float)C;
    mv[0] = m; mv[1] = rsqrtf(var + 1e-5f);
  }
  __syncthreads();
  if (t < C) x[row * C + t] = fmaxf((v - mv[0]) * mv[1] * g[t] + bta[t], 0.f);
}

// ================================================================ launch
extern "C" void kernel_launch(void* const* d_in, const int* in_sizes, int n_in,
                              void* d_out, int out_size, void* d_ws, size_t ws_size,
                              hipStream_t stream) {
  const float* xyz  = (const float*)d_in[0];
  const float* goal = (const float*)d_in[1];
  int i = 2;
  // ---- sa1 params (insertion-order flatten of nested dicts)
  const float* s1_mlp_w  = (const float*)d_in[i++];
  const float* s1_mlp_b  = (const float*)d_in[i++];
  const float* s1_mlp_g  = (const float*)d_in[i++];
  const float* s1_mlp_be = (const float*)d_in[i++];
  WnP wn1;
  wn1.w0 = (const float*)d_in[i++]; wn1.b0 = (const float*)d_in[i++];
  wn1.g0 = (const float*)d_in[i++]; wn1.be0 = (const float*)d_in[i++];
  wn1.w1 = (const float*)d_in[i++]; wn1.b1 = (const float*)d_in[i++];
  wn1.g1 = (const float*)d_in[i++]; wn1.be1 = (const float*)d_in[i++];
  wn1.w2 = (const float*)d_in[i++]; wn1.b2 = (const float*)d_in[i++];
  wn1.g2 = (const float*)d_in[i++]; wn1.be2 = (const float*)d_in[i++];
  const float* s1_lin_w  = (const float*)d_in[i++];
  const float* s1_lin_b  = (const float*)d_in[i++];
  const float* s1_lin_g  = (const float*)d_in[i++];
  const float* s1_lin_be = (const float*)d_in[i++];
  // ---- sa3 params
  const float* s3_mlp_w  = (const float*)d_in[i++];
  const float* s3_mlp_b  = (const float*)d_in[i++];
  const float* s3_mlp_g  = (const float*)d_in[i++];
  const float* s3_mlp_be = (const float*)d_in[i++];
  WnP wn3;
  wn3.w0 = (const float*)d_in[i++]; wn3.b0 = (const float*)d_in[i++];
  wn3.g0 = (const float*)d_in[i++]; wn3.be0 = (const float*)d_in[i++];
  wn3.w1 = (const float*)d_in[i++]; wn3.b1 = (const float*)d_in[i++];
  wn3.g1 = (const float*)d_in[i++]; wn3.be1 = (const float*)d_in[i++];
  wn3.w2 = (const float*)d_in[i++]; wn3.b2 = (const float*)d_in[i++];
  wn3.g2 = (const float*)d_in[i++]; wn3.be2 = (const float*)d_in[i++];
  const float* s3_lin_w  = (const float*)d_in[i++];
  const float* s3_lin_b  = (const float*)d_in[i++];
  const float* s3_lin_g  = (const float*)d_in[i++];
  const float* s3_lin_be = (const float*)d_in[i++];
  // ---- head
  const float* fc1_w = (const float*)d_in[i++];
  const float* fc1_b = (const float*)d_in[i++];
  const float* gn1_g = (const float*)d_in[i++];
  const float* gn1_b = (const float*)d_in[i++];
  const float* fc3_w = (const float*)d_in[i++];
  const float* fc3_b = (const float*)d_in[i++];
  const float* gn3_g = (const float*)d_in[i++];
  const float* gn3_b = (const float*)d_in[i++];
  const float* fc4_w = (const float*)d_in[i++];
  const float* fc4_b = (const float*)d_in[i++];
  const float* gn4_g = (const float*)d_in[i++];
  const float* gn4_b = (const float*)d_in[i++];
  const float* fc5_w = (const float*)d_in[i++];
  const float* fc5_b = (const float*)d_in[i++];

  // ---- workspace carve-up (256B aligned)
  char* wp = (char*)d_ws;
  auto alloc = [&](size_t nbytes) {
    char* p = wp; wp += (nbytes + 255) & ~(size_t)255; return p;
  };
  float* sqn   = (float*)alloc((size_t)BB * NPTS * 4);
  float* invd  = (float*)alloc((size_t)BB * NPTS * 4);
  int*   fpsi  = (int*)  alloc((size_t)BB * NP1 * 4);
  int*   knni  = (int*)  alloc((size_t)BB * NP1 * 8 * 4);
  float* l1x   = (float*)alloc((size_t)BB * NP1 * 3 * 4);
  float* A1    = (float*)alloc((size_t)BB * NP1 * 1024 * 4);
  float* l1p   = (float*)alloc((size_t)BB * NP1 * 64 * 4);
  float* invd3 = (float*)alloc((size_t)BB * NP1 * 4);
  float* A3    = (float*)alloc((size_t)BB * 2048 * 4);
  float* l3    = (float*)alloc((size_t)BB * 128 * 4);
  float* hbuf  = (float*)alloc((size_t)16 * 128 * 4);
  float* h1    = (float*)alloc((size_t)16 * 64 * 4);
  float* h3    = (float*)alloc((size_t)16 * 32 * 4);
  float* h4    = (float*)alloc((size_t)16 * 16 * 4);
  float* out   = (float*)d_out;   // (16, 3)

  // 1) point squared norms
  sqnorm_kernel<<<(BB * NPTS + 255) / 256, 256, 0, stream>>>(xyz, goal, sqn, NPTS);
  // 2) Gaussian density via WMMA pairwise GEMM (bw=0.1)
  density_kernel<<<BB * (NPTS / 16), 32, 0, stream>>>(xyz, goal, sqn, invd, NPTS, 0.1f);
  // 3) farthest point sampling (64 points)
  fps_kernel<<<BB, 256, 0, stream>>>(xyz, goal, fpsi, NPTS, NP1);
  // 4) kNN (8) around sampled centers; also emits l1_xyz
  knn_kernel<<<BB * NP1, 64, 0, stream>>>(xyz, goal, fpsi, knni, l1x, NPTS);
  // 5) SA1 grouping + MLP + density-scale + weightnet + einsum -> A1 (BB,64,1024)
  sa1_group_kernel<<<BB * NP1, 64, 0, stream>>>(xyz, goal, knni, invd, l1x,
                                                s1_mlp_w, s1_mlp_b, s1_mlp_g, s1_mlp_be,
                                                wn1, A1, NPTS);
  // 6) SA1 linear (WMMA, per-batch 64x64x1024) + bias + bn + relu -> l1_pts (BB,64,64)
  gemm_wmma_kernel<<<dim3(4, 4, BB), 32, 0, stream>>>(
      A1, s1_lin_w, s1_lin_b, s1_lin_g, s1_lin_be, l1p,
      64, 64, 1024, (long long)64 * 1024, (long long)64 * 64, 1);
  // 7) SA3 density (bw=0.4) over the 64 sampled points
  density3_kernel<<<BB, 64, 0, stream>>>(l1x, invd3);
  // 8) SA3 fused stage -> A3 (BB, 2048)
  sa3_kernel<<<BB, 128, 0, stream>>>(l1x, l1p, invd3,
                                     s3_mlp_w, s3_mlp_b, s3_mlp_g, s3_mlp_be, wn3, A3);
  // 9) SA3 linear (WMMA, 32x128x2048) + bias + bn + relu -> l3 (BB,128)
  gemm_wmma_kernel<<<dim3(128 / 16, 32 / 16, 1), 32, 0, stream>>>(
      A3, s3_lin_w, s3_lin_b, s3_lin_g, s3_lin_be, l3,
      BB, 128, 2048, 0, 0, 1);
  // 10) h = encode(goal) - encode(xyz)
  diff_kernel<<<(16 * 128 + 255) / 256, 256, 0, stream>>>(l3, hbuf);
  // 11) fc1 (16x64x128) -> groupnorm+relu
  gemm_wmma_kernel<<<dim3(4, 1, 1), 32, 0, stream>>>(
      hbuf, fc1_w, fc1_b, nullptr, nullptr, h1, 16, 64, 128, 0, 0, 0);
  groupnorm_relu_kernel<<<16, 64, 0, stream>>>(h1, gn1_g, gn1_b, 64);
  // 12) fc3 (16x32x64) -> gn+relu
  gemm_wmma_kernel<<<dim3(2, 1, 1), 32, 0, stream>>>(
      h1, fc3_w, fc3_b, nullptr, nullptr, h3, 16, 32, 64, 0, 0, 0);
  groupnorm_relu_kernel<<<16, 64, 0, stream>>>(h3, gn3_g, gn3_b, 32);
  // 13) fc4 (16x16x32) -> gn+relu
  gemm_wmma_kernel<<<dim3(1, 1, 1), 32, 0, stream>>>(
      h3, fc4_w, fc4_b, nullptr, nullptr, h4, 16, 16, 32, 0, 0, 0);
  groupnorm_relu_kernel<<<16, 64, 0, stream>>>(h4, gn4_g, gn4_b, 16);
  // 14) fc5 (16x3x16) -> output
  gemm_wmma_kernel<<<dim3(1, 1, 1), 32, 0, stream>>>(
      h4, fc5_w, fc5_b, nullptr, nullptr, out, 16, 3, 16, 0, 0, 0);
  (void)in_sizes; (void)n_in; (void)out_size; (void)ws_size;
}